// TrialPsi_90469191123169
// MI455X (gfx1250) — compile-verified
//
#include <hip/hip_runtime.h>
#include <math.h>
#include <stdint.h>

// ---------------------------------------------------------------------------
// VMC Metropolis walk, MI455X (gfx1250, wave32).
//
// Roofline: 256 MB of (noise, unif) streamed once @ 23.3 TB/s => ~11 us floor.
// Math simplified so each measured step needs 1 exp + 1 rcp + 1 log + ~35 VALU
// per walker; 11 real accumulators, complex constants applied in an epilogue.
//
// Data path: per-wave 8-deep LDS ring filled with global_load_async_to_lds_b32
// (ASYNCcnt), gated by s_wait_asynccnt — CDNA5 async-copy path, no VGPR return,
// no cross-wave sync (each wave consumes only its own slabs).
// ---------------------------------------------------------------------------

#define NW      65536
#define TOTALT  512
#define WARM    128
#define NSTEPS  384
#define TPB     256
#define DEPTH   8              // prefetch depth (stages per wave)
#define STAGE_B 256            // bytes per wave per stage: 32 noise + 32 unif floats
#define SMEM_B  (8 * DEPTH * STAGE_B)   // 8 waves/block * 8 stages * 256B = 16 KB

__global__ __launch_bounds__(TPB)
void vmc_walk_kernel(const float* __restrict__ theta,
                     const float* __restrict__ r0,
                     const float* __restrict__ s_dev_p,
                     const float* __restrict__ noise,
                     const float* __restrict__ unif,
                     float* __restrict__ ws)
{
    extern __shared__ float dyn[];   // SMEM_B bytes; staging ring, then reused for reduction

    const uint32_t tid    = threadIdx.x;
    const uint32_t lane   = tid & 31u;
    const uint32_t wave   = tid >> 5;
    const uint32_t walker = blockIdx.x * TPB + tid;   // 1 walker per thread

    const float th1  = theta[1];
    const float th2  = theta[2];
    const float sdev = s_dev_p[0];
    const float m2th1 = -2.0f * th1;
    const float p2th1 =  2.0f * th1;

    // LDS byte offset of dyn[] : flat LDS addresses carry the LDS offset in
    // their low 32 bits (ISA 10.2 aperture truncation), so this matches what
    // the hardware uses for both the async writes and the ds reads.
    const uint32_t lds_base  = (uint32_t)(uintptr_t)(void*)dyn;
    const uint32_t wave_slab = lds_base + wave * (DEPTH * STAGE_B);

    // ---- prologue: fill the ring (2 async loads per stage, in-order) -------
    for (int s = 0; s < DEPTH; ++s) {
        uint32_t dstN = wave_slab + (uint32_t)s * STAGE_B + lane * 4u;
        uint32_t dstU = dstN + 128u;
        uint32_t voff = (uint32_t)(s * NW + (int)walker) * 4u;   // byte offset, < 2^31
        asm volatile("global_load_async_to_lds_b32 %0, %1, %2"
                     :: "v"(dstN), "v"(voff), "s"(noise) : "memory");
        asm volatile("global_load_async_to_lds_b32 %0, %1, %2"
                     :: "v"(dstU), "v"(voff), "s"(unif)  : "memory");
    }

    float r = r0[walker];
    float x = r - th2;
    float x2 = x * x;

    float A0=0.f,A1=0.f,A2=0.f,A3=0.f,A4=0.f,A5=0.f,
          A6=0.f,A7=0.f,A8=0.f,A9=0.f,A10=0.f;

    for (int t = 0; t < TOTALT; ++t) {
        const uint32_t stage  = (uint32_t)t & (DEPTH - 1u);
        // Oldest pair (current stage) complete when <= 2*(DEPTH-1) outstanding.
        asm volatile("s_wait_asynccnt 14" ::: "memory");

        const uint32_t slab_f = (wave * DEPTH + stage) * (STAGE_B / 4);
        float n = dyn[slab_f + lane];         // ds_load_b32
        float u = dyn[slab_f + 32u + lane];   // ds_load_b32

        // Metropolis: accept if log(u) < -2*th1*((rp-th2)^2 - (r-th2)^2)
        float rp   = fmaf(sdev, n, r);
        float xp   = rp - th2;
        float xp2  = xp * xp;
        float lrat = m2th1 * (xp2 - x2);
        bool  acc  = (__logf(u) < lrat);
        r  = acc ? rp  : r;
        x  = acc ? xp  : x;
        x2 = acc ? xp2 : x2;

        if (t >= WARM) {
            // e = C*cosh(a) + i*S*sinh(a), a = 2*th1*x; C,S applied in epilogue
            float a  = p2th1 * x;
            float ep = __expf(a);
            float em = __builtin_amdgcn_rcpf(ep);
            float ch = 0.5f * (ep + em);
            float sh = 0.5f * (ep - em);
            A0 += ch;  A1 += sh;  A2 += r;  A3 += x2;  A4 += x;
            A5  = fmaf(r,  sh, A5);   // -> EDs0.re (* S)
            A6  = fmaf(r,  ch, A6);   // -> EDs0.im (* -C)
            A7  = fmaf(ch, x2, A7);   // -> EDs1.re (* C)
            A8  = fmaf(sh, x2, A8);   // -> EDs1.im (* S)
            A9  = fmaf(ch, x,  A9);   // -> EDs2.re (* 2*th1*C)
            A10 = fmaf(sh, x,  A10);  // -> EDs2.im (* 2*th1*S)
        }

        // Our ds reads of this stage must land before the refill overwrites it
        // (async LDS writes are unordered vs ds reads).
        asm volatile("s_wait_dscnt 0" ::: "memory");

        // Refill consumed stage with step t+DEPTH (clamped at the tail so the
        // outstanding count stays constant and the immediate wait stays valid;
        // the dummy refill only touches the stage we just finished consuming).
        int tn = t + DEPTH; if (tn > TOTALT - 1) tn = TOTALT - 1;
        uint32_t dstN = wave_slab + stage * STAGE_B + lane * 4u;
        uint32_t dstU = dstN + 128u;
        uint32_t voff = (uint32_t)(tn * NW + (int)walker) * 4u;
        asm volatile("global_load_async_to_lds_b32 %0, %1, %2"
                     :: "v"(dstN), "v"(voff), "s"(noise) : "memory");
        asm volatile("global_load_async_to_lds_b32 %0, %1, %2"
                     :: "v"(dstU), "v"(voff), "s"(unif)  : "memory");
    }

    // Drain all async writes before repurposing the staging LDS.
    asm volatile("s_wait_asynccnt 0" ::: "memory");
    __syncthreads();

    // ---- deterministic block reduction (11 sums, fixed order) --------------
    float A[11] = {A0,A1,A2,A3,A4,A5,A6,A7,A8,A9,A10};
    #pragma unroll
    for (int k = 0; k < 11; ++k) dyn[k * TPB + tid] = A[k];
    __syncthreads();
    for (int stride = TPB / 2; stride > 0; stride >>= 1) {
        if (tid < (uint32_t)stride) {
            #pragma unroll
            for (int k = 0; k < 11; ++k)
                dyn[k * TPB + tid] += dyn[k * TPB + tid + stride];
        }
        __syncthreads();
    }
    if (tid < 11u) ws[blockIdx.x * 11 + tid] = dyn[tid * TPB];
}

// Epilogue: sum per-block partials in fixed order, apply complex constants,
// write the 14 output floats (Es, Ds[3], EDs[3] as re/im pairs).
__global__ void vmc_finalize_kernel(const float* __restrict__ theta,
                                    const float* __restrict__ ws,
                                    float* __restrict__ out,
                                    int nblocks)
{
    __shared__ float S[11];
    const int k = threadIdx.x;
    if (k < 11) {
        float s = 0.0f;
        for (int b = 0; b < nblocks; ++b) s += ws[b * 11 + k];
        S[k] = s;
    }
    __syncthreads();
    if (k == 0) {
        const float th0 = theta[0], th1 = theta[1];
        const float t_hop = 1.0f;                       // T
        const float emt = __expf(-th1);
        const float C  = -2.0f * t_hop * emt * __cosf(th0);
        const float Sg =  2.0f * t_hop * emt * __sinf(th0);
        const float p2th1 = 2.0f * th1;
        const float scale = 1.0f / ((float)NW * (float)NSTEPS);
        out[0]  = C  * S[0] * scale;            // Es.re
        out[1]  = Sg * S[1] * scale;            // Es.im
        out[2]  = 0.0f;                         // Ds[0].re (i*r is imaginary)
        out[3]  = S[2] * scale;                 // Ds[0].im = mean(r)
        out[4]  = S[3] * scale;                 // Ds[1].re = mean(x^2)
        out[5]  = 0.0f;                         // Ds[1].im
        out[6]  = p2th1 * S[4] * scale;         // Ds[2].re = mean(2*th1*x)
        out[7]  = 0.0f;                         // Ds[2].im
        out[8]  =  Sg * S[5] * scale;           // EDs[0].re =  S*mean(r*sh)
        out[9]  = -C  * S[6] * scale;           // EDs[0].im = -C*mean(r*ch)
        out[10] =  C  * S[7] * scale;           // EDs[1].re
        out[11] =  Sg * S[8] * scale;           // EDs[1].im
        out[12] = p2th1 * C  * S[9]  * scale;   // EDs[2].re
        out[13] = p2th1 * Sg * S[10] * scale;   // EDs[2].im
    }
}

extern "C" void kernel_launch(void* const* d_in, const int* in_sizes, int n_in,
                              void* d_out, int out_size, void* d_ws, size_t ws_size,
                              hipStream_t stream) {
    (void)in_sizes; (void)n_in; (void)out_size; (void)ws_size;
    const float* theta = (const float*)d_in[0];
    const float* r0    = (const float*)d_in[1];
    const float* sdev  = (const float*)d_in[2];
    const float* noise = (const float*)d_in[3];
    const float* unif  = (const float*)d_in[4];
    float* out = (float*)d_out;
    float* ws  = (float*)d_ws;           // needs 256*11*4 = 11264 bytes

    const int nblocks = NW / TPB;        // 256 blocks * 256 threads = 65536 walkers
    hipLaunchKernelGGL(vmc_walk_kernel, dim3(nblocks), dim3(TPB), SMEM_B, stream,
                       theta, r0, sdev, noise, unif, ws);
    hipLaunchKernelGGL(vmc_finalize_kernel, dim3(1), dim3(32), 0, stream,
                       theta, ws, out, nblocks);
}